// BloomStageLoss_25194278159144
// MI455X (gfx1250) — compile-verified
//
#include <hip/hip_runtime.h>
#include <stdint.h>

#ifndef __has_builtin
#define __has_builtin(x) 0
#endif

#define BLOCK        256
#define NBLK_MAX     4096
#define TILE_ROWS    256
#define TILE_FLOATS  (TILE_ROWS * 5)   // 1280 floats = 5120 bytes per tile

// ---------- CDNA5 async global->LDS path (guarded; falls back to sync copy) ----
#if __has_builtin(__builtin_amdgcn_global_load_async_to_lds_b128) && \
    __has_builtin(__builtin_amdgcn_global_load_async_to_lds_b32)
#define HAVE_ASYNC_LDS 1
#else
#define HAVE_ASYNC_LDS 0
#endif

// Builtin parameter types (confirmed by hipcc diagnostics over two rounds):
//   b128: (v4i __device__ *src_global, v4i __shared__ *dst_lds, int off, int cpol)
//   b32 : (int __device__ *src_global, int __shared__ *dst_lds, int off, int cpol)
typedef int v4i_t __attribute__((vector_size(16)));            // matches builtin V4i
typedef __attribute__((address_space(1))) v4i_t* gptr_v4i;     // AS1 = __device__
typedef __attribute__((address_space(1))) int*   gptr_i32;
typedef __attribute__((address_space(3))) v4i_t* lptr_v4i;     // AS3 = __shared__
typedef __attribute__((address_space(3))) int*   lptr_i32;

__device__ __forceinline__ gptr_v4i as_g_v4i(const void* p) {
  return (gptr_v4i)(uintptr_t)p;
}
__device__ __forceinline__ gptr_i32 as_g_i32(const void* p) {
  return (gptr_i32)(uintptr_t)p;
}
__device__ __forceinline__ lptr_v4i as_l_v4i(void* p) {
  return (lptr_v4i)(uint32_t)(uintptr_t)p;  // low 32 bits of generic = LDS offset
}
__device__ __forceinline__ lptr_i32 as_l_i32(void* p) {
  return (lptr_i32)(uint32_t)(uintptr_t)p;
}

#if HAVE_ASYNC_LDS
__device__ __forceinline__ void wait_async_0() {
#if __has_builtin(__builtin_amdgcn_s_wait_asynccnt)
  __builtin_amdgcn_s_wait_asynccnt(0);
#else
  asm volatile("s_wait_asynccnt 0x0" ::: "memory");
#endif
}
__device__ __forceinline__ void wait_async_2() {   // 2 newest ops = next tile in flight
#if __has_builtin(__builtin_amdgcn_s_wait_asynccnt)
  __builtin_amdgcn_s_wait_asynccnt(2);
#else
  asm volatile("s_wait_asynccnt 0x2" ::: "memory");
#endif
}
#endif

// Issue one tile (1280 floats): 256 lanes x b128 covers floats [0,1024),
// 256 lanes x b32 covers [1024,1280). Exactly 2 async ops per lane per tile.
__device__ __forceinline__ void issue_tile_load(const float* __restrict__ x,
                                                int tile, float* buf, int tid) {
  const float* g = x + (size_t)tile * TILE_FLOATS;
#if HAVE_ASYNC_LDS
  __builtin_amdgcn_global_load_async_to_lds_b128(as_g_v4i(g + tid * 4),
                                                 as_l_v4i(buf + tid * 4), 0, 0);
  __builtin_amdgcn_global_load_async_to_lds_b32(as_g_i32(g + 1024 + tid),
                                                as_l_i32(buf + 1024 + tid), 0, 0);
#else
  const float4 v = *(const float4*)(g + tid * 4);
  *(float4*)(buf + tid * 4) = v;
  buf[1024 + tid] = g[1024 + tid];
#endif
}

// ---------- fast math helpers (TRANS32 hardware ops) ---------------------------
__device__ __forceinline__ float fast_exp2(float a) {
#if __has_builtin(__builtin_amdgcn_exp2f)
  return __builtin_amdgcn_exp2f(a);
#else
  return exp2f(a);
#endif
}
__device__ __forceinline__ float fast_log2(float a) {
#if __has_builtin(__builtin_amdgcn_logf)
  return __builtin_amdgcn_logf(a);
#else
  return log2f(a);
#endif
}
__device__ __forceinline__ float fast_rcp(float a) {
#if __has_builtin(__builtin_amdgcn_rcpf)
  return __builtin_amdgcn_rcpf(a);
#else
  return 1.0f / a;
#endif
}

// Transition weight T[t][j]: d=|t-j| -> 0, 0.5, 1.0, 2.0, 2.0
// = (d==0) ? 0 : min(2, 2^(d-2))   -- one v_ldexp + v_min + v_cndmask
__device__ __forceinline__ float tdist_w(int t, int j) {
  int d = t - j; if (d < 0) d = -d;
#if __has_builtin(__builtin_amdgcn_ldexpf)
  float w = __builtin_amdgcn_ldexpf(1.0f, d - 2);
#else
  float w = ldexpf(1.0f, (int)(d - 2));
#endif
  w = fminf(w, 2.0f);
  return (d == 0) ? 0.0f : w;
}

// Per-row loss. CE folded: weights are 0.9 on target, 0.025 elsewhere, sum to 1,
// so  ce = (m + lnS) - 0.875*x_t - 0.025*sum(x).
__device__ __forceinline__ float row_loss(float x0, float x1, float x2,
                                          float x3, float x4, int t) {
  const float m  = fmaxf(fmaxf(fmaxf(x0, x1), fmaxf(x2, x3)), x4);
  const float L2E = 1.44269504088896340736f;  // log2(e)
  const float e0 = fast_exp2((x0 - m) * L2E);
  const float e1 = fast_exp2((x1 - m) * L2E);
  const float e2 = fast_exp2((x2 - m) * L2E);
  const float e3 = fast_exp2((x3 - m) * L2E);
  const float e4 = fast_exp2((x4 - m) * L2E);
  const float S   = ((e0 + e1) + (e2 + e3)) + e4;
  const float lnS = 0.69314718055994530942f * fast_log2(S);
  const float sumx = ((x0 + x1) + (x2 + x3)) + x4;
  const float xt = (t == 0) ? x0 : (t == 1) ? x1 : (t == 2) ? x2 : (t == 3) ? x3 : x4;
  const float ce = (m + lnS) - fmaf(0.875f, xt, 0.025f * sumx);

  float ws = e0 * tdist_w(t, 0);
  ws = fmaf(e1, tdist_w(t, 1), ws);
  ws = fmaf(e2, tdist_w(t, 2), ws);
  ws = fmaf(e3, tdist_w(t, 3), ws);
  ws = fmaf(e4, tdist_w(t, 4), ws);
  return ce + 0.1f * (ws * fast_rcp(S));
}

// ---------- kernel 1: per-block partial sums ----------------------------------
__global__ __launch_bounds__(BLOCK) void bloom_loss_partial(
    const float* __restrict__ x, const int* __restrict__ tgt,
    float* __restrict__ partial, int B, int nTilesFull) {
  __shared__ __align__(16) float tileBuf[2][TILE_FLOATS];  // 10 KB double buffer
  __shared__ float red[BLOCK];

  const int tid = threadIdx.x;
  const int stride = gridDim.x;
  float acc = 0.0f;

  int tile = blockIdx.x;
  int cur = 0;
  if (tile < nTilesFull) issue_tile_load(x, tile, tileBuf[0], tid);

  while (tile < nTilesFull) {
    const int nxt = tile + stride;
    const bool hasNext = nxt < nTilesFull;
    if (hasNext) issue_tile_load(x, nxt, tileBuf[cur ^ 1], tid);  // overlap next
#if HAVE_ASYNC_LDS
    if (hasNext) wait_async_2(); else wait_async_0();  // current tile resident
#endif
    __syncthreads();

    // prefetch tile two strides ahead into L2 (global_prefetch_b8)
    {
      const long long pf = (long long)tile + 2LL * stride;
      if (pf < (long long)nTilesFull && tid < (TILE_FLOATS * 4) / 128) {
        __builtin_prefetch((const char*)x + (size_t)pf * TILE_FLOATS * 4 +
                               (size_t)tid * 128, 0, 1);
      }
    }

    {  // stride-5 LDS read: gcd(5,64)=1 -> bank-conflict-free
      const float* r = &tileBuf[cur][tid * 5];
      const float v0 = r[0], v1 = r[1], v2 = r[2], v3 = r[3], v4 = r[4];
      const int t = tgt[(size_t)tile * TILE_ROWS + tid];
      acc += row_loss(v0, v1, v2, v3, v4, t);
    }
    __syncthreads();  // all waves done reading before this buffer is reloaded
    cur ^= 1;
    tile = nxt;
  }

  // tail rows (B % 256), handled by one designated block straight from global
  const int tailStart = nTilesFull * TILE_ROWS;
  if (tailStart < B && (int)blockIdx.x == (nTilesFull % stride)) {
    const int row = tailStart + tid;
    if (row < B) {
      const float* r = x + (size_t)row * 5;
      acc += row_loss(r[0], r[1], r[2], r[3], r[4], tgt[row]);
    }
  }

  red[tid] = acc;
  __syncthreads();
  for (int s = BLOCK / 2; s > 0; s >>= 1) {
    if (tid < s) red[tid] += red[tid + s];
    __syncthreads();
  }
  if (tid == 0) partial[blockIdx.x] = red[0];
}

// ---------- kernel 2: deterministic finalize (f64 accumulate) -----------------
__global__ __launch_bounds__(BLOCK) void bloom_loss_finalize(
    const float* __restrict__ partial, int n, float* __restrict__ out, int B) {
  __shared__ double red[BLOCK];
  double a = 0.0;
  for (int i = threadIdx.x; i < n; i += BLOCK) a += (double)partial[i];
  red[threadIdx.x] = a;
  __syncthreads();
  for (int s = BLOCK / 2; s > 0; s >>= 1) {
    if ((int)threadIdx.x < s) red[threadIdx.x] += red[threadIdx.x + s];
    __syncthreads();
  }
  if (threadIdx.x == 0) out[0] = (float)(red[0] / (double)B);
}

extern "C" void kernel_launch(void* const* d_in, const int* in_sizes, int n_in,
                              void* d_out, int out_size, void* d_ws, size_t ws_size,
                              hipStream_t stream) {
  const float* x  = (const float*)d_in[0];   // [B,5] f32
  const int* tgt  = (const int*)d_in[1];     // [B]   i32
  const int B = in_sizes[1];

  int nblk = NBLK_MAX;
  if (ws_size < (size_t)nblk * sizeof(float)) nblk = (int)(ws_size / sizeof(float));
  if (nblk < 1) nblk = 1;

  const int nTilesFull = B / TILE_ROWS;
  float* partial = (float*)d_ws;

  bloom_loss_partial<<<nblk, BLOCK, 0, stream>>>(x, tgt, partial, B, nTilesFull);
  bloom_loss_finalize<<<1, BLOCK, 0, stream>>>(partial, nblk, (float*)d_out, B);
}